// synthetic_dnn_ae_v1_quantized_32057635898018
// MI455X (gfx1250) — compile-verified
//
#include <hip/hip_runtime.h>
#include <hip/hip_bf16.h>
#include <math.h>

// ---------------------------------------------------------------------------
// Fused quantized autoencoder, MI455X / gfx1250.
//   encoder: 32->20 -> qtanh(16,2) -> 20->14 -> qtanh(16,2) -> 14->3
//   decoder: qtanh(16,14) -> 3->14 -> qtanh(16,2) -> 14->20 -> qtanh(16,2) -> 20->32
// Strategy:
//   * setup kernel folds BN into quantized weights, pre-swizzles them into the
//     V_WMMA_F32_16X16X4_F32 B-fragment lane layout, and tabulates the 33-point
//     step-tanh (exact: reference only ever evaluates tanh on that grid).
//   * main kernel: 1 wave == 16 batch rows. Activations live in a wave-private
//     LDS tile (stride 36 floats -> conflict-free, 16B aligned). All 6 layers
//     run as chained f32 WMMAs (44 per tile), epilogues (fxp + LUT tanh) on the
//     8-VGPR accumulators, outputs streamed straight to HBM.
// Roofline: 268 MB mandatory HBM traffic @ 23.3 TB/s ~= 11.5 us floor; ~2.9M
// f32 WMMAs and the LUT-based epilogue are far below that -> memory-bound.
// ---------------------------------------------------------------------------

typedef __attribute__((ext_vector_type(2))) float v2f;
typedef __attribute__((ext_vector_type(8))) float v8f;

#if defined(__HIP_DEVICE_COMPILE__)
#if !__has_builtin(__builtin_amdgcn_wmma_f32_16x16x4_f32)
#error "wmma_f32_16x16x4_f32 builtin not available on gfx1250 toolchain"
#endif
#endif

// ---- fixed-point helpers (match jnp.round == round-half-even) ---------------
__device__ __forceinline__ float fxp_q(float x, float scale, float invscale) {
  float r = rintf(x * scale);
  r = fminf(fmaxf(r, -32768.0f), 32767.0f);
  return r * invscale;
}
__device__ __forceinline__ float fxp14(float x) { return fxp_q(x, 16384.0f, 6.103515625e-05f); }
__device__ __forceinline__ float fxp2(float x)  { return fxp_q(x, 4.0f, 0.25f); }

// step-tanh via LUT: idx = rint(clamp(x,-3,3)*31/6) + 16  in [0,32]
__device__ __forceinline__ float qtanh_lut(float x, const float* __restrict__ lut) {
  const float inv_step = 31.0f / 6.0f;
  int idx = (int)rintf(fminf(fmaxf(x, -3.0f), 3.0f) * inv_step) + 16;
  return lut[idx];
}

// ---- layout constants -------------------------------------------------------
#define N_LAYERS 6
#define ACT_STRIDE 36   // floats; stride 36 -> LDS conflict-free & 16B aligned
// frag area (lane-major pairs):
//   L0 0..1023, L1 1024..1343, L2 1344..1599, L3 1600..1663, L4 1664..2175, L5 2176..2815
// biases at 2816 + {0,32,48,64,80,112} (Npad each), LUTs at 2960 (fxp2) / 2993 (fxp14)
#define LUT2_OFF  2960
#define LUT14_OFF 2993
#define WS_FLOATS 3026
#define SMEM_FLOATS (WS_FLOATS + 8 * 16 * ACT_STRIDE)

enum { ACT_HID = 0, ACT_ENC = 1, ACT_DEC = 2 };

// ---------------------------------------------------------------------------
// Per-layer fused WMMA block. A from wave-private LDS act tile, B from the
// pre-swizzled weight fragments in LDS. C/D layout: lane (0-15|16-31) holds
// N = lane&15, rows M = r + 8*(lane>>4) in accumulator regs r=0..7.
// ---------------------------------------------------------------------------
template <int FI, int FO, int KPAD, int NT, int FOFF, int BOFF, int MODE>
__device__ __forceinline__ void run_layer(const float* __restrict__ smemW,
                                          float* __restrict__ act, int lane,
                                          float* __restrict__ outEnc,
                                          float* __restrict__ outDec, int rowbase) {
  constexpr int KS = KPAD / 4;
  const int g  = lane >> 4;
  const int ln = lane & 15;
  v8f c[NT];
#pragma unroll
  for (int nt = 0; nt < NT; ++nt) {
    c[nt] = (v8f){0.f, 0.f, 0.f, 0.f, 0.f, 0.f, 0.f, 0.f};
#pragma unroll
    for (int ks = 0; ks < KS; ++ks) {
      // A fragment: 32-bit A 16x4 layout -> lane holds M=ln, K = ksbase + r + 2g
      const int k0 = ks * 4 + 2 * g;
      v2f a;
      a.x = act[ln * ACT_STRIDE + k0];
      a.y = act[ln * ACT_STRIDE + k0 + 1];
      // B fragment: pre-swizzled lane-major pair
      const float* f = smemW + FOFF + ((nt * KS + ks) * 32 + lane) * 2;
      v2f b;
      b.x = f[0];
      b.y = f[1];
      c[nt] = __builtin_amdgcn_wmma_f32_16x16x4_f32(false, a, false, b, (short)0,
                                                    c[nt], false, false);
    }
  }
  // Epilogue: bias + fxp(16,14) [+ store] [+ LUT step-tanh], write act tile.
  const float* lut = smemW + (MODE == ACT_ENC ? LUT14_OFF : LUT2_OFF);
#pragma unroll
  for (int nt = 0; nt < NT; ++nt) {
    const int N = nt * 16 + ln;
    const float bias = smemW[BOFF + N];
    float* actp = act + (8 * g) * ACT_STRIDE + N;                 // + r*ACT_STRIDE
    float* encp = outEnc + (size_t)(rowbase + 8 * g) * FO + N;    // + r*FO
    float* decp = outDec + (size_t)(rowbase + 8 * g) * 32 + N;    // + r*32
#pragma unroll
    for (int r = 0; r < 8; ++r) {
      float v = c[nt][r] + bias;
      v = fxp14(v);
      if (MODE == ACT_ENC) {
        if (ln < FO) encp[r * FO] = v;
        v = qtanh_lut(v, lut);              // decoder input: qtanh(16,14)
      } else if (MODE == ACT_DEC) {
        decp[r * 32] = v;
      } else {
        v = qtanh_lut(v, lut);              // hidden: qtanh(16,2)
      }
      if (MODE != ACT_DEC) actp[r * ACT_STRIDE] = (N < FO) ? v : 0.0f;
    }
  }
}

// ---------------------------------------------------------------------------
// Setup kernel: fold BN into quantized weights/bias, emit WMMA B fragments,
// tabulate the 33-point step-tanh for both output quantizations.
// ---------------------------------------------------------------------------
struct LayerParams { const float *w, *b, *g, *be, *rm, *rv; };
struct AllParams { LayerParams p[N_LAYERS]; };

__global__ void ae_setup(AllParams P, float* __restrict__ ws) {
  const int FI[N_LAYERS]   = {32, 20, 14, 3, 14, 20};
  const int FO[N_LAYERS]   = {20, 14, 3, 14, 20, 32};
  const int NT[N_LAYERS]   = {2, 1, 1, 1, 2, 2};
  const int KS[N_LAYERS]   = {8, 5, 4, 1, 4, 5};
  const int FOFF[N_LAYERS] = {0, 1024, 1344, 1600, 1664, 2176};
  const int BOFF[N_LAYERS] = {2816, 2848, 2864, 2880, 2896, 2928};
  const int NPAD[N_LAYERS] = {32, 16, 16, 16, 32, 32};

  for (int l = 0; l < N_LAYERS; ++l) {
    const LayerParams& q = P.p[l];
    const int nfrag = NT[l] * KS[l] * 64;
    for (int i = threadIdx.x; i < nfrag; i += blockDim.x) {
      const int r    = i & 1;
      const int lane = (i >> 1) & 31;
      const int step = i >> 6;                 // nt*KS + ks
      const int nt = step / KS[l], ks = step % KS[l];
      const int K = ks * 4 + 2 * (lane >> 4) + r;
      const int N = nt * 16 + (lane & 15);
      float val = 0.0f;
      if (K < FI[l] && N < FO[l]) {
        const float s = q.g[N] * rsqrtf(q.rv[N] + 1e-5f);
        val = fxp14(q.w[N * FI[l] + K]) * s;
      }
      ws[FOFF[l] + i] = val;
    }
    for (int N = threadIdx.x; N < NPAD[l]; N += blockDim.x) {
      float val = 0.0f;
      if (N < FO[l]) {
        const float s = q.g[N] * rsqrtf(q.rv[N] + 1e-5f);
        val = (fxp14(q.b[N]) - q.rm[N]) * s + q.be[N];
      }
      ws[BOFF[l] + N] = val;
    }
  }
  // step-tanh LUTs: xs = (i-16) * 6/31, i = 0..32
  for (int i = threadIdx.x; i < 33; i += blockDim.x) {
    const float xs = (float)(i - 16) * (6.0f / 31.0f);
    const float t  = tanhf(xs);
    ws[LUT2_OFF + i]  = fxp2(t);
    ws[LUT14_OFF + i] = fxp14(t);
  }
}

// ---------------------------------------------------------------------------
// Main fused kernel: 256 threads = 8 waves, each wave owns one 16-row tile.
// ---------------------------------------------------------------------------
__global__ __launch_bounds__(256) void ae_main(const float* __restrict__ x,
                                               const float* __restrict__ wsbuf,
                                               float* __restrict__ outEnc,
                                               float* __restrict__ outDec,
                                               int ntiles) {
  __shared__ float smem[SMEM_FLOATS];
  const int tid = threadIdx.x;
  for (int i = tid; i < WS_FLOATS; i += 256) smem[i] = wsbuf[i];
  __syncthreads();

  const int wave = tid >> 5;
  const int lane = tid & 31;
  float* act = smem + WS_FLOATS + wave * (16 * ACT_STRIDE);

  const int gwave  = blockIdx.x * 8 + wave;
  const int nwaves = gridDim.x * 8;

  for (int tile = gwave; tile < ntiles; tile += nwaves) {
    const int rowbase = tile << 4;
    // prefetch next tile's input (speculative; OOB is dropped)
    __builtin_prefetch(x + (size_t)(tile + nwaves) * 512, 0, 1);

    // Stage 16x32 f32 input tile into LDS, fully coalesced float4 loads.
    const float4* src = reinterpret_cast<const float4*>(x + (size_t)rowbase * 32);
#pragma unroll
    for (int p = 0; p < 4; ++p) {
      const int idx  = p * 32 + lane;       // index into 128 float4s
      const float4 v = src[idx];
      const int row = (idx * 4) >> 5;
      const int col = (idx * 4) & 31;
      *reinterpret_cast<float4*>(&act[row * ACT_STRIDE + col]) = v;
    }
    // wave-private tile: per-wave LDS ordering guarantees store->load order.

    run_layer<32, 20, 32, 2,    0, 2816, ACT_HID>(smem, act, lane, outEnc, outDec, rowbase);
    run_layer<20, 14, 20, 1, 1024, 2848, ACT_HID>(smem, act, lane, outEnc, outDec, rowbase);
    run_layer<14,  3, 16, 1, 1344, 2864, ACT_ENC>(smem, act, lane, outEnc, outDec, rowbase);
    run_layer< 3, 14,  4, 1, 1600, 2880, ACT_HID>(smem, act, lane, outEnc, outDec, rowbase);
    run_layer<14, 20, 16, 2, 1664, 2896, ACT_HID>(smem, act, lane, outEnc, outDec, rowbase);
    run_layer<20, 32, 20, 2, 2176, 2928, ACT_DEC>(smem, act, lane, outEnc, outDec, rowbase);
  }
}

// ---------------------------------------------------------------------------
extern "C" void kernel_launch(void* const* d_in, const int* in_sizes, int n_in,
                              void* d_out, int out_size, void* d_ws, size_t ws_size,
                              hipStream_t stream) {
  (void)in_sizes; (void)n_in; (void)out_size; (void)ws_size;
  const float* x = (const float*)d_in[0];
  AllParams P;
  for (int l = 0; l < N_LAYERS; ++l) {
    P.p[l].w  = (const float*)d_in[1 + l * 6 + 0];
    P.p[l].b  = (const float*)d_in[1 + l * 6 + 1];
    P.p[l].g  = (const float*)d_in[1 + l * 6 + 2];
    P.p[l].be = (const float*)d_in[1 + l * 6 + 3];
    P.p[l].rm = (const float*)d_in[1 + l * 6 + 4];
    P.p[l].rv = (const float*)d_in[1 + l * 6 + 5];
  }
  float* ws = (float*)d_ws;
  ae_setup<<<1, 256, 0, stream>>>(P, ws);

  float* outEnc = (float*)d_out;
  float* outDec = outEnc + (size_t)1048576 * 3;
  const int ntiles = 1048576 / 16;          // 65536 tiles of 16 rows
  ae_main<<<2048, 256, 0, stream>>>(x, ws, outEnc, outDec, ntiles);
}